// HadamardProj_89223650607577
// MI455X (gfx1250) — compile-verified
//
#include <hip/hip_runtime.h>

typedef __attribute__((ext_vector_type(2))) float v2f;
typedef __attribute__((ext_vector_type(8))) float v8f;

__device__ __forceinline__ float hsignf(int a, int b) {
    // (-1)^popcount(a&b) as float: parity goes straight into the sign bit.
    // No v_cmp / v_cndmask, no VCC serialization.
    unsigned s = ((unsigned)__popc(a & b)) << 31;   // bit0 -> bit31
    return __uint_as_float(0x3F800000u ^ s);
}

__device__ __forceinline__ v8f wmma_f32_4(v2f a, v2f b, v8f c) {
    return __builtin_amdgcn_wmma_f32_16x16x4_f32(
        false, a, false, b, (short)0, c, false, false);
}

// One wave32 processes rows in a grid-stride loop:
//   row (1024 f32) viewed as X[32][32];  Y = H32 * X * H32   (H1024 = H32 (x) H32)
//   out[j] = Y[j/32][j%32] * (weight / (||x|| * 32)) + bias[j],  j < 1000
__global__ __launch_bounds__(256) void hadamard_proj_wmma(
    const float* __restrict__ x,
    const float* __restrict__ weight,
    const float* __restrict__ bias,
    float* __restrict__ out,
    int batch)
{
    const int  lane   = threadIdx.x & 31;
    const bool hi     = lane >= 16;
    const int  l16    = lane & 15;
    const int  kofs   = hi ? 2 : 0;
    const int  waveId = (blockIdx.x * blockDim.x + threadIdx.x) >> 5;
    const int  nWaves = (gridDim.x * blockDim.x) >> 5;

    const float wscale = weight[0] * 0.03125f;   // weight / sqrt(1024)

    // ---- Hoisted H32 operand tiles. H32 is symmetric, so the WMMA A-layout
    //      and B-layout register images coincide: H[tt][ks] serves as
    //      B(ks, nt=tt) in matmul 1 and A(mt=tt, ks) in matmul 2.
    v2f H[2][8];
#pragma unroll
    for (int tt = 0; tt < 2; ++tt) {
        const int t = tt * 16 + l16;
#pragma unroll
        for (int ks = 0; ks < 8; ++ks) {
            const int k = ks * 4 + kofs;
            H[tt][ks][0] = hsignf(k,     t);
            H[tt][ks][1] = hsignf(k + 1, t);
        }
    }

    // ---- Hoisted bias: per-lane output index j is row-invariant.
    float bv[2][2][8];
#pragma unroll
    for (int mt = 0; mt < 2; ++mt)
#pragma unroll
        for (int nt = 0; nt < 2; ++nt)
#pragma unroll
            for (int rr = 0; rr < 8; ++rr) {
                const int j = (mt * 16 + rr + (hi ? 8 : 0)) * 32 + nt * 16 + l16;
                bv[mt][nt][rr] = (j < 1000) ? bias[j] : 0.0f;
            }

    for (int row = waveId; row < batch; row += nWaves) {   // wave-uniform loop
        const float* __restrict__ xrow = x + (size_t)row * 1024;

        // ---- Single pass load of X in WMMA A-layout (contiguous 8B per lane),
        //      accumulating sum of squares on the fly.
        v2f  Xa[2][8];
        float sumsq = 0.0f;
#pragma unroll
        for (int mt = 0; mt < 2; ++mt) {
            const int m = mt * 16 + l16;
#pragma unroll
            for (int ks = 0; ks < 8; ++ks) {
                const int k = ks * 4 + kofs;
                v2f v = *(const v2f*)(xrow + m * 32 + k);
                Xa[mt][ks] = v;
                sumsq += v[0] * v[0] + v[1] * v[1];
            }
        }
#pragma unroll
        for (int off = 16; off > 0; off >>= 1)
            sumsq += __shfl_xor(sumsq, off, 32);
        const float s = rsqrtf(sumsq) * wscale;   // applied at the end (linear)

        // ---- W = X * H32   (B tile H[nt][ks] shared across both mt accumulators)
        v8f W[2][2];
#pragma unroll
        for (int nt = 0; nt < 2; ++nt) {
            v8f a0 = {0.f, 0.f, 0.f, 0.f, 0.f, 0.f, 0.f, 0.f};
            v8f a1 = {0.f, 0.f, 0.f, 0.f, 0.f, 0.f, 0.f, 0.f};
#pragma unroll
            for (int ks = 0; ks < 8; ++ks) {
                a0 = wmma_f32_4(Xa[0][ks], H[nt][ks], a0);
                a1 = wmma_f32_4(Xa[1][ks], H[nt][ks], a1);
            }
            W[0][nt] = a0;
            W[1][nt] = a1;
        }

        // ---- Y = H32 * W.  B tile = W slice converted C/D-layout -> B-layout
        //      (one shfl_xor(16) + one select per element pair), computed once
        //      per (nt, ks) and fed to both mt accumulators.
        v8f Y[2][2];
#pragma unroll
        for (int nt = 0; nt < 2; ++nt) {
            v8f a0 = {0.f, 0.f, 0.f, 0.f, 0.f, 0.f, 0.f, 0.f};
            v8f a1 = {0.f, 0.f, 0.f, 0.f, 0.f, 0.f, 0.f, 0.f};
#pragma unroll
            for (int ks = 0; ks < 8; ++ks) {
                const v8f wt  = W[ks >> 2][nt];   // W rows 0-15 / 16-31
                const int ksl = ks & 3;
                v2f Bv;
#pragma unroll
                for (int v = 0; v < 2; ++v) {
                    const int klo = ksl * 4 + v;  // K row needed by low lanes
                    const int khi = klo + 2;      // K row needed by high lanes
                    float vlo = (klo < 8) ? wt[klo] : __shfl_xor(wt[klo - 8], 16, 32);
                    float vhi = (khi < 8) ? __shfl_xor(wt[khi], 16, 32) : wt[khi - 8];
                    Bv[v] = hi ? vhi : vlo;
                }
                a0 = wmma_f32_4(H[0][ks], Bv, a0);
                a1 = wmma_f32_4(H[1][ks], Bv, a1);
            }
            Y[0][nt] = a0;
            Y[1][nt] = a1;
        }

        // ---- scale, add (pre-loaded) bias, store first 1000 of 1024 outputs
        float* __restrict__ orow = out + (size_t)row * 1000;
#pragma unroll
        for (int mt = 0; mt < 2; ++mt) {
#pragma unroll
            for (int nt = 0; nt < 2; ++nt) {
#pragma unroll
                for (int rr = 0; rr < 8; ++rr) {
                    const int M = mt * 16 + rr + (hi ? 8 : 0);
                    const int N = nt * 16 + l16;
                    const int j = M * 32 + N;
                    if (j < 1000)
                        orow[j] = Y[mt][nt][rr] * s + bv[mt][nt][rr];
                }
            }
        }
    }
}

extern "C" void kernel_launch(void* const* d_in, const int* in_sizes, int n_in,
                              void* d_out, int out_size, void* d_ws, size_t ws_size,
                              hipStream_t stream) {
    const float* x      = (const float*)d_in[0];
    const float* weight = (const float*)d_in[1];
    const float* bias   = (const float*)d_in[2];
    float*       out    = (float*)d_out;

    const int batch = in_sizes[0] / 1024;       // 16384

    // 512 blocks x 8 waves = 4096 waves -> 4 rows per wave at batch=16384.
    // Hoisted H/bias registers amortize across the per-wave row loop.
    int blocks = (batch + 31) / 32;             // ceil(batch / (8 waves * 4 rows))
    if (blocks > 512) blocks = 512;
    if (blocks < 1)   blocks = 1;

    hadamard_proj_wmma<<<blocks, 256, 0, stream>>>(x, weight, bias, out, batch);
}